// BettiMatchingLoss_28458453303849
// MI455X (gfx1250) — compile-verified
//
#include <hip/hip_runtime.h>
#include <stdint.h>

// Problem constants (from the reference)
constexpr int Din = 128, Hin = 256, Win = 256;     // input  (B=4, C=1, D, H, W)
constexpr int OD  = 64,  OH  = 128, OW  = 128;     // output (B=4, C=1, OD, OH, OW)
constexpr int NUM_OUT_ROWS = 4 * OD * OH;          // 32768 output rows of 128 elems
constexpr long OUT_PER_TENSOR = (long)4 * OD * OH * OW;  // 4,194,304

// gfx1250 async global->LDS copy, 16 bytes per lane. ldsOff is a byte offset
// into LDS (low 32 bits of the generic shared-aperture address). Tracked by
// ASYNCcnt. th:TH_LOAD_NT: pure streaming read, no reuse (302MB > 192MB L2).
__device__ __forceinline__ void async_copy_b128(uint32_t ldsOff, const void* gaddr) {
    asm volatile("global_load_async_to_lds_b128 %0, %1, off th:TH_LOAD_NT"
                 :: "v"(ldsOff), "v"(gaddr)
                 : "memory");
}

__device__ __forceinline__ void wait_asynccnt0() {
    asm volatile("s_wait_asynccnt 0" ::: "memory");
}

__global__ __launch_bounds__(256) void
betti_prep_kernel(const float* __restrict__ logits,
                  const float* __restrict__ target,
                  float* __restrict__ out)
{
    // 16 input rows of 256 floats: [tensor(2)][outRowHalf(2)][sub(4)][W(256)]
    __shared__ float smem[16 * 256];

    const int tid = threadIdx.x;
    const int blk = blockIdx.x;

    const uint32_t smemBase = (uint32_t)(uintptr_t)(&smem[0]); // LDS byte offset

    // ---- Stage 1: async-stage 16 KB (16 rows x 1 KB) into LDS ------------
    // 1024 b128 transfers total, 4 per thread.
#pragma unroll
    for (int i = 0; i < 4; ++i) {
        const int t      = tid + (i << 8);  // transfer id 0..1023
        const int row    = t >> 6;          // 0..15  (which staged row)
        const int chunk  = t & 63;          // 16B chunk within the 1KB row
        const int tensor = row >> 3;        // 0 = logits, 1 = target
        const int half   = (row >> 2) & 1;  // which of the block's 2 out rows
        const int sub    = row & 3;         // d_off*2 + h_off

        const int r  = 2 * blk + half;      // global output row id
        const int b  = r >> 13;             // / (OD*OH) = / 8192
        const int od = (r >> 7) & 63;
        const int oh = r & 127;

        const int di = 2 * od + (sub >> 1);
        const int hi = 2 * oh + (sub & 1);

        const float* src = tensor ? target : logits;
        const float* g = src + (((size_t)b * Din + di) * Hin + hi) * Win
                             + (chunk << 2);

        const uint32_t ldsOff = smemBase + (uint32_t)((row << 10) + (chunk << 4));
        async_copy_b128(ldsOff, g);
    }

    wait_asynccnt0();       // this wave's async transfers complete
    __syncthreads();        // make all waves' staged data visible

    // ---- Stage 2: 2x2x2 max-pool + pointwise from LDS --------------------
    const int half = tid >> 7;      // which output row of the pair
    const int ow   = tid & 127;     // output column
    const int r    = 2 * blk + half;

    float m  = -__builtin_inff();   // pooled logit max
    float mt = -__builtin_inff();   // pooled target max
#pragma unroll
    for (int sub = 0; sub < 4; ++sub) {
        const float2 a = *reinterpret_cast<const float2*>(
            &smem[((half * 4 + sub) << 8) + 2 * ow]);          // logits rows 0..7
        const float2 c = *reinterpret_cast<const float2*>(
            &smem[((8 + half * 4 + sub) << 8) + 2 * ow]);      // target rows 8..15
        m  = fmaxf(m,  fmaxf(a.x, a.y));
        mt = fmaxf(mt, fmaxf(c.x, c.y));
    }

    // pred_super = 1 - sigmoid(max) = sigmoid(-max) = 1/(1+exp(max))
    const float pred = 1.0f / (1.0f + expf(m));
    // target_super = 1 - (max > 0.5)
    const float tgt  = (mt > 0.5f) ? 0.0f : 1.0f;

    const size_t o = (size_t)r * OW + ow;
    out[o] = pred;
    out[OUT_PER_TENSOR + o] = tgt;
}

extern "C" void kernel_launch(void* const* d_in, const int* in_sizes, int n_in,
                              void* d_out, int out_size, void* d_ws, size_t ws_size,
                              hipStream_t stream) {
    const float* logits = (const float*)d_in[0];
    const float* target = (const float*)d_in[1];
    float* out = (float*)d_out;

    const int blocks = NUM_OUT_ROWS / 2;   // 16384 blocks, 2 output rows each
    betti_prep_kernel<<<blocks, 256, 0, stream>>>(logits, target, out);
}